// DGCNN_88716844466195
// MI455X (gfx1250) — compile-verified
//
#include <hip/hip_runtime.h>
#include <math.h>

typedef float v2f __attribute__((ext_vector_type(2)));
typedef float v8f __attribute__((ext_vector_type(8)));

#define KNN_K 20
#define NPTS  4096
#define NB    4
#define PPIX  (NPTS * KNN_K)   // 81920 pixels per batch

// ---------------------------------------------------------------------------
// 1) KNN: for each query point, top-20 by largest -||d||^2 (ties -> lower idx)
// ---------------------------------------------------------------------------
__global__ __launch_bounds__(256)
void knn_kernel(const float* __restrict__ pts, int* __restrict__ idx)
{
    const int N = NPTS;
    const int b = blockIdx.y;
    const int n = blockIdx.x * 256 + threadIdx.x;
    const float* pb = pts + (size_t)b * 6 * N;

    const float qx = pb[n], qy = pb[N + n], qz = pb[2 * N + n];

    float best[KNN_K];
    int   bidx[KNN_K];
#pragma unroll
    for (int i = 0; i < KNN_K; i++) { best[i] = -3.402823466e38f; bidx[i] = 0; }

    __shared__ float sx[256], sy[256], sz[256];

    for (int t = 0; t < N; t += 256) {
        const int m = t + threadIdx.x;
        sx[threadIdx.x] = pb[m];
        sy[threadIdx.x] = pb[N + m];
        sz[threadIdx.x] = pb[2 * N + m];
        __syncthreads();
        for (int j = 0; j < 256; j++) {
            const float dx = qx - sx[j], dy = qy - sy[j], dz = qz - sz[j];
            const float val = -(dx * dx + dy * dy + dz * dz);
            if (val > best[KNN_K - 1]) {
                int pos = KNN_K - 1;
                while (pos > 0 && best[pos - 1] < val) {  // strict: equal keeps earlier idx first
                    best[pos] = best[pos - 1]; bidx[pos] = bidx[pos - 1]; pos--;
                }
                best[pos] = val; bidx[pos] = t + j;
            }
        }
        __syncthreads();
    }

    int* op = idx + ((size_t)b * N + n) * KNN_K;
#pragma unroll
    for (int i = 0; i < KNN_K; i++) op[i] = bidx[i];
}

// ---------------------------------------------------------------------------
// 2) Graph feature: 16 channels (13 real + 3 zero-pad so CIN % 4 == 0)
//    X layout (B, 16, P) pixel-major -> GEMM-ready.
// ---------------------------------------------------------------------------
__device__ __forceinline__ float angle3(float ax, float ay, float az,
                                        float bx, float by, float bz)
{
    const float cx = ay * bz - az * by;
    const float cy = az * bx - ax * bz;
    const float cz = ax * by - ay * bx;
    const float cn = sqrtf(cx * cx + cy * cy + cz * cz);
    const float d  = ax * bx + ay * by + az * bz;
    return atan2f(cn, d);
}

__global__ __launch_bounds__(256)
void feat_kernel(const float* __restrict__ pts, const int* __restrict__ idx,
                 float* __restrict__ X)
{
    const int N = NPTS;
    const int t = blockIdx.x * 256 + threadIdx.x;
    if (t >= NB * N * KNN_K) return;

    const int kc = t % KNN_K;
    const int n  = (t / KNN_K) % N;
    const int b  = t / (KNN_K * N);

    const float* pb = pts + (size_t)b * 6 * N;
    const int m = idx[t];   // t == ((b*N+n)*K + kc)

    const float cx = pb[n],          cy = pb[N + n],      cz = pb[2 * N + n];
    const float gx = pb[m],          gy = pb[N + m],      gz = pb[2 * N + m];
    const float lx = gx - cx,        ly = gy - cy,        lz = gz - cz;
    const float nrx = pb[3 * N + n], nry = pb[4 * N + n], nrz = pb[5 * N + n];
    const float nix = pb[3 * N + m], niy = pb[4 * N + m], niz = pb[5 * N + m];

    const float nr_d  = angle3(nrx, nry, nrz, lx, ly, lz);
    const float ni_d  = angle3(nix, niy, niz, lx, ly, lz);
    const float nr_ni = angle3(nrx, nry, nrz, nix, niy, niz);
    const float dn    = sqrtf(lx * lx + ly * ly + lz * lz);

    const size_t P = PPIX;
    const size_t base = (size_t)b * 16 * P + (size_t)n * KNN_K + kc;
    X[base +  0 * P] = gx;   X[base +  1 * P] = gy;   X[base +  2 * P] = gz;
    X[base +  3 * P] = cx;   X[base +  4 * P] = cy;   X[base +  5 * P] = cz;
    X[base +  6 * P] = lx;   X[base +  7 * P] = ly;   X[base +  8 * P] = lz;
    X[base +  9 * P] = nr_d; X[base + 10 * P] = ni_d; X[base + 11 * P] = nr_ni;
    X[base + 12 * P] = dn;
    X[base + 13 * P] = 0.0f; X[base + 14 * P] = 0.0f; X[base + 15 * P] = 0.0f;
}

// Pad W1 (64x13) -> W1p (64x16) with zero columns.
__global__ __launch_bounds__(256)
void padw1_kernel(const float* __restrict__ W1, float* __restrict__ W1p)
{
    const int t = blockIdx.x * 256 + threadIdx.x;
    if (t >= 64 * 16) return;
    const int o = t / 16, c = t % 16;
    W1p[t] = (c < 13) ? W1[o * 13 + c] : 0.0f;
}

// ---------------------------------------------------------------------------
// 3) GEMM via V_WMMA_F32_16X16X4_F32, compile-time shapes, branch-free loop.
//    Y[b,o,p] = sum_c W[o,c] * act(X[b,c,p]); act = prev-layer GN affine+ReLU.
//    Wave tile: M=32 x N=64. W tile + affine staged in LDS (pitch CIN+4:
//    half-wave banks {4l,4l+1} vs {4l+2,4l+3} -> conflict-free b64 loads).
// ---------------------------------------------------------------------------
template <int CIN, int COUT, bool AFF>
__global__ __launch_bounds__(256)
void gemm_wmma_kernel(const float* __restrict__ X, const float* __restrict__ Wm,
                      const float* __restrict__ aff, float* __restrict__ Y)
{
    constexpr int P   = PPIX;
    constexpr int LDW = CIN + 4;

    __shared__ float sW[32 * LDW];
    __shared__ float sAff[AFF ? 2 * CIN : 1];

    const int tid  = threadIdx.x;
    const int lane = tid & 31;
    const int wave = tid >> 5;
    const int half = lane >> 4;        // 0: K={c0,c0+1}, 1: K={c0+2,c0+3}
    const int lr   = lane & 15;
    const int b    = blockIdx.z;
    const int m0   = blockIdx.y * 32;
    const int p0   = (blockIdx.x * 8 + wave) * 64;

    // Stage W rows m0..m0+31 (contiguous in memory) into LDS, coalesced.
    {
        const float* wsrc = Wm + (size_t)m0 * CIN;
        for (int i = tid; i < 32 * CIN; i += 256)
            sW[(i / CIN) * LDW + (i % CIN)] = wsrc[i];
        if (AFF) {
            const float* ab = aff + (size_t)b * 2 * CIN;
            for (int i = tid; i < 2 * CIN; i += 256) sAff[i] = ab[i];
        }
    }
    __syncthreads();

    const float* Xb = X + (size_t)b * CIN * P;
    v8f acc[2][4] = {};

    for (int c0 = 0; c0 < CIN; c0 += 4) {
        const int ka = c0 + half * 2;

        v2f a0, a1;
        a0.x = sW[lr * LDW + ka];
        a0.y = sW[lr * LDW + ka + 1];
        a1.x = sW[(16 + lr) * LDW + ka];
        a1.y = sW[(16 + lr) * LDW + ka + 1];

        float s0 = 1.0f, s1 = 1.0f, o0 = 0.0f, o1 = 0.0f;
        if (AFF) {
            s0 = sAff[ka];     o0 = sAff[CIN + ka];
            s1 = sAff[ka + 1]; o1 = sAff[CIN + ka + 1];
        }

#pragma unroll
        for (int nt = 0; nt < 4; nt++) {
            const float* xp = Xb + (size_t)ka * P + (p0 + nt * 16 + lr);
            float x0 = xp[0];
            float x1 = xp[P];
            if (AFF) {
                x0 = fmaxf(fmaf(s0, x0, o0), 0.0f);   // GroupNorm affine + ReLU
                x1 = fmaxf(fmaf(s1, x1, o1), 0.0f);
            }
            v2f bf; bf.x = x0; bf.y = x1;
            acc[0][nt] = __builtin_amdgcn_wmma_f32_16x16x4_f32(
                false, a0, false, bf, (short)0, acc[0][nt], false, false);
            acc[1][nt] = __builtin_amdgcn_wmma_f32_16x16x4_f32(
                false, a1, false, bf, (short)0, acc[1][nt], false, false);
        }
    }

    // C/D layout: VGPR r, lanes 0-15 -> (M=r, N=lane); lanes 16-31 -> (M=r+8)
#pragma unroll
    for (int mt = 0; mt < 2; mt++) {
#pragma unroll
        for (int nt = 0; nt < 4; nt++) {
            const int p = p0 + nt * 16 + lr;
#pragma unroll
            for (int r = 0; r < 8; r++) {
                const int m = m0 + mt * 16 + r + half * 8;
                Y[((size_t)b * COUT + m) * P + p] = acc[mt][nt][r];
            }
        }
    }
}

// ---------------------------------------------------------------------------
// 4) GroupNorm stats: one block per (group, batch); deterministic tree
//    reduction; emits fused per-channel scale/bias for the consumer.
// ---------------------------------------------------------------------------
__global__ __launch_bounds__(256)
void stats_kernel(const float* __restrict__ Y, const float* __restrict__ gamma,
                  const float* __restrict__ beta, float* __restrict__ aff,
                  int C)
{
    const int P  = PPIX;
    const int g  = blockIdx.x;
    const int b  = blockIdx.y;
    const int Cg = C / 16;     // GROUPS = 16
    const float* Yb = Y + ((size_t)b * C + (size_t)g * Cg) * P;

    double s = 0.0, ss = 0.0;
    for (int c = 0; c < Cg; c++) {
        const float* row = Yb + (size_t)c * P;
        for (int p = threadIdx.x; p < P; p += 256) {
            const float v = row[p];
            s += v; ss += (double)v * v;
        }
    }

    __shared__ double rs[256], rss[256];
    rs[threadIdx.x] = s; rss[threadIdx.x] = ss;
    __syncthreads();
    for (int off = 128; off > 0; off >>= 1) {
        if (threadIdx.x < off) {
            rs[threadIdx.x]  += rs[threadIdx.x + off];
            rss[threadIdx.x] += rss[threadIdx.x + off];
        }
        __syncthreads();
    }

    __shared__ float smean, srstd;
    if (threadIdx.x == 0) {
        const double cnt  = (double)Cg * (double)P;
        const double mean = rs[0] / cnt;
        const double var  = rss[0] / cnt - mean * mean;
        smean = (float)mean;
        srstd = rsqrtf((float)var + 1e-5f);
    }
    __syncthreads();

    if (threadIdx.x < Cg) {
        const int c = g * Cg + threadIdx.x;
        const float scl = gamma[c] * srstd;
        aff[(size_t)b * 2 * C + c]     = scl;                     // scale
        aff[(size_t)b * 2 * C + C + c] = beta[c] - smean * scl;   // bias
    }
}

// ---------------------------------------------------------------------------
// 5) Final: out[b,o,n] = max_k relu(scale*Y4 + bias)
// ---------------------------------------------------------------------------
__global__ __launch_bounds__(256)
void final_kernel(const float* __restrict__ Y, const float* __restrict__ aff,
                  float* __restrict__ out)
{
    const int C = 256, N = NPTS;
    const int t = blockIdx.x * 256 + threadIdx.x;
    if (t >= NB * C * N) return;
    const int n = t % N;
    const int o = (t / N) % C;
    const int b = t / (N * C);

    const float scl = aff[(size_t)b * 2 * C + o];
    const float ofs = aff[(size_t)b * 2 * C + C + o];
    const float* yp = Y + (((size_t)b * C + o) * N + n) * KNN_K;

    float mx = 0.0f;   // ReLU outputs are >= 0
#pragma unroll
    for (int k = 0; k < KNN_K; k++)
        mx = fmaxf(mx, fmaxf(fmaf(scl, yp[k], ofs), 0.0f));
    out[t] = mx;
}

// ---------------------------------------------------------------------------
extern "C" void kernel_launch(void* const* d_in, const int* in_sizes, int n_in,
                              void* d_out, int out_size, void* d_ws, size_t ws_size,
                              hipStream_t stream)
{
    const int B = NB, N = NPTS, P = PPIX;

    const float* pts = (const float*)d_in[0];
    const float* W1 = (const float*)d_in[1];
    const float* g1 = (const float*)d_in[2];
    const float* b1 = (const float*)d_in[3];
    const float* W2 = (const float*)d_in[4];
    const float* g2 = (const float*)d_in[5];
    const float* b2 = (const float*)d_in[6];
    const float* W3 = (const float*)d_in[7];
    const float* g3 = (const float*)d_in[8];
    const float* b3 = (const float*)d_in[9];
    const float* W4 = (const float*)d_in[10];
    const float* g4 = (const float*)d_in[11];
    const float* b4 = (const float*)d_in[12];
    float* out = (float*)d_out;

    // Workspace layout (float elements). Y4 aliases the X0/Y1/Y2 region,
    // which is dead by the time layer 4 runs.
    float* ws = (float*)d_ws;
    const size_t oX0  = 0;                               // 16*P*B
    const size_t oY1  = (size_t)16 * P * B;              // 64*P*B
    const size_t oY2  = oY1 + (size_t)64 * P * B;        // 64*P*B
    const size_t oY3  = (size_t)256 * P * B;             // after Y4's region
    const size_t oY4  = 0;
    const size_t oIdx = oY3 + (size_t)128 * P * B;
    const size_t oW1p = oIdx + (size_t)B * N * KNN_K;
    const size_t oA1  = oW1p + 64 * 16;
    const size_t oA2  = oA1 + (size_t)B * 2 * 64;
    const size_t oA3  = oA2 + (size_t)B * 2 * 64;
    const size_t oA4  = oA3 + (size_t)B * 2 * 128;

    float* X0  = ws + oX0;  float* Y1 = ws + oY1;  float* Y2 = ws + oY2;
    float* Y3  = ws + oY3;  float* Y4 = ws + oY4;
    int*   idxb = (int*)(ws + oIdx);
    float* W1p = ws + oW1p;
    float* A1  = ws + oA1;  float* A2 = ws + oA2;
    float* A3  = ws + oA3;  float* A4 = ws + oA4;

    // 1) KNN + graph features (16-channel, zero-padded)
    knn_kernel<<<dim3(N / 256, B), 256, 0, stream>>>(pts, idxb);
    feat_kernel<<<(B * N * KNN_K + 255) / 256, 256, 0, stream>>>(pts, idxb, X0);
    padw1_kernel<<<(64 * 16 + 255) / 256, 256, 0, stream>>>(W1, W1p);

    const int px = P / (8 * 64);   // 160 pixel-tiles per grid.x
    // 2) Layer 1: 16 -> 64 (raw features, no affine)
    gemm_wmma_kernel<16, 64, false><<<dim3(px, 64 / 32, B), 256, 0, stream>>>(X0, W1p, A1, Y1);
    stats_kernel<<<dim3(16, B), 256, 0, stream>>>(Y1, g1, b1, A1, 64);
    // 3) Layer 2: 64 -> 64 (fused GN+ReLU of layer 1 on input)
    gemm_wmma_kernel<64, 64, true><<<dim3(px, 64 / 32, B), 256, 0, stream>>>(Y1, W2, A1, Y2);
    stats_kernel<<<dim3(16, B), 256, 0, stream>>>(Y2, g2, b2, A2, 64);
    // 4) Layer 3: 64 -> 128
    gemm_wmma_kernel<64, 128, true><<<dim3(px, 128 / 32, B), 256, 0, stream>>>(Y2, W3, A2, Y3);
    stats_kernel<<<dim3(16, B), 256, 0, stream>>>(Y3, g3, b3, A3, 128);
    // 5) Layer 4: 128 -> 256
    gemm_wmma_kernel<128, 256, true><<<dim3(px, 256 / 32, B), 256, 0, stream>>>(Y3, W4, A3, Y4);
    stats_kernel<<<dim3(16, B), 256, 0, stream>>>(Y4, g4, b4, A4, 256);
    // 6) GN+ReLU+max over K
    final_kernel<<<(B * 256 * N + 255) / 256, 256, 0, stream>>>(Y4, A4, out);
}